// LocalSpatialEncoding_43696997269970
// MI455X (gfx1250) — compile-verified
//
#include <hip/hip_runtime.h>

typedef float v2f __attribute__((ext_vector_type(2)));
typedef float v8f __attribute__((ext_vector_type(8)));

#define BSZ 4
#define NSZ 32768
#define KSZ 16
#define DSZ 8
#define NLOG 15
#define EPSV 1e-6f

// Build the per-lane B-matrix fragments for W^T (10->16 cols, cols 8..15 zero).
// B layout (4x16, 2 VGPRs): VGPR v, lanes 0-15 -> row v, lanes 16-31 -> row v+2.
__device__ __forceinline__ void make_bmats(const float* __restrict__ W, int lane,
                                           v2f& bm0, v2f& bm1, v2f& bm2) {
    const int col = lane & 15;
    const int hi  = lane >> 4;
    const bool valid = (col < DSZ);
    const float* wr = W + col * 10;
    float w0 = 0.f, w1 = 0.f, w2 = 0.f, w3 = 0.f, w4 = 0.f, w5 = 0.f;
    if (valid) {
        // channel rows this lane owns: chunk j, rows j*4 + (hi?2:0) + {0,1}
        w0 = wr[hi ? 2 : 0];  w1 = wr[hi ? 3 : 1];   // chunk0: ch 0..3
        w2 = wr[hi ? 6 : 4];  w3 = wr[hi ? 7 : 5];   // chunk1: ch 4..7
        w4 = hi ? 0.f : wr[8];                       // chunk2: ch 8,9 (10,11 pad)
        w5 = hi ? 0.f : wr[9];
    }
    bm0 = v2f{w0, w1};
    bm1 = v2f{w2, w3};
    bm2 = v2f{w4, w5};
}

// Compute x = concat(10ch) @ W^T for one (b,n): 16 neighbors x 16 channels.
// Returns D fragment: lane c (lo) holds channel c for k=0..7 in v[0..7];
// hi lanes hold k=8..15 for channel (lane-16).
__device__ __forceinline__ v8f compute_x(const float* __restrict__ coords,
                                         const int*   __restrict__ idx,
                                         const float* __restrict__ dist,
                                         int b, int n, int lane,
                                         v2f bm0, v2f bm1, v2f bm2) {
    const int k  = lane & 15;        // A-matrix row M = neighbor slot
    const int hi = lane >> 4;
    const int base_nk = (b * NSZ + n) * KSZ + k;
    const int j = idx[base_nk];
    const float dd = dist[base_nk];
    const float* pc = coords + (size_t)(b * NSZ + n) * 3;
    const float* pn = coords + (size_t)(b * NSZ + j) * 3;
    const float cx = pc[0], cy = pc[1], cz = pc[2];
    const float nx = pn[0], ny = pn[1], nz = pn[2];
    const float rx = cx - nx, ry = cy - ny, rz = cz - nz;
    // concat channels: 0..2=center 3..5=neighbor 6..8=relative 9=dist 10,11=pad
    // A layout: lo lanes hold K pair {0,1} of each chunk, hi lanes pair {2,3}.
    v2f a0 = v2f{hi ? cz : cx, hi ? nx : cy};
    v2f a1 = v2f{hi ? rx : ny, hi ? ry : nz};
    v2f a2 = v2f{hi ? 0.f : rz, hi ? 0.f : dd};
    v8f acc = {};
    acc = __builtin_amdgcn_wmma_f32_16x16x4_f32(false, a0, false, bm0, (short)0, acc, false, false);
    acc = __builtin_amdgcn_wmma_f32_16x16x4_f32(false, a1, false, bm1, (short)0, acc, false, false);
    acc = __builtin_amdgcn_wmma_f32_16x16x4_f32(false, a2, false, bm2, (short)0, acc, false, false);
    return acc;
}

__global__ void lse_zero_ws(float* __restrict__ ws) {
    if (threadIdx.x < 2 * DSZ) ws[threadIdx.x] = 0.f;
}

// Pass 1: per-channel sum and sum-of-squares of x (conv output incl. bias).
__global__ void lse_stats(const float* __restrict__ coords,
                          const int*   __restrict__ idx,
                          const float* __restrict__ dist,
                          const float* __restrict__ W,
                          const float* __restrict__ bias,
                          float* __restrict__ ws) {
    __shared__ float s_sum[DSZ];
    __shared__ float s_sq[DSZ];
    const int tid = threadIdx.x;
    if (tid < DSZ) { s_sum[tid] = 0.f; s_sq[tid] = 0.f; }
    __syncthreads();

    const int lane = tid & 31;
    const int col  = lane & 15;
    v2f bm0, bm1, bm2;
    make_bmats(W, lane, bm0, bm1, bm2);
    const float bb = (col < DSZ) ? bias[col] : 0.f;

    const int wave   = (blockIdx.x * blockDim.x + tid) >> 5;
    const int nwaves = (gridDim.x * blockDim.x) >> 5;

    float s = 0.f, sq = 0.f;
    for (int row = wave; row < BSZ * NSZ; row += nwaves) {
        const int b = row >> NLOG;
        const int n = row & (NSZ - 1);
        v8f x = compute_x(coords, idx, dist, b, n, lane, bm0, bm1, bm2);
#pragma unroll
        for (int r = 0; r < 8; ++r) {
            const float v = x[r] + bb;
            s  += v;
            sq += v * v;
        }
    }
    if (col < DSZ) {
        atomicAdd(&s_sum[col], s);
        atomicAdd(&s_sq[col],  sq);
    }
    __syncthreads();
    if (tid < DSZ) {
        atomicAdd(&ws[tid],        s_sum[tid]);
        atomicAdd(&ws[DSZ + tid],  s_sq[tid]);
    }
}

// Pass 2: recompute x via WMMA, batchnorm+relu, write [encoded | features].
// One wave per (b,n); each lane stores 8 contiguous floats (coalesced).
__global__ void lse_main(const float* __restrict__ coords,
                         const float* __restrict__ features,
                         const int*   __restrict__ idx,
                         const float* __restrict__ dist,
                         const float* __restrict__ W,
                         const float* __restrict__ bias,
                         const float* __restrict__ gamma,
                         const float* __restrict__ beta,
                         const float* __restrict__ ws,
                         float* __restrict__ out) {
    const int gtid = blockIdx.x * blockDim.x + threadIdx.x;
    const int wave = gtid >> 5;
    const int lane = threadIdx.x & 31;
    const int b = wave >> NLOG;
    const int n = wave & (NSZ - 1);
    const int col = lane & 15;
    const int hi  = lane >> 4;

    v2f bm0, bm1, bm2;
    make_bmats(W, lane, bm0, bm1, bm2);
    v8f x = compute_x(coords, idx, dist, b, n, lane, bm0, bm1, bm2);

    float vals[8];
    if (col < DSZ) {
        const float cnt  = (float)((long long)BSZ * NSZ * KSZ);
        const float mean = ws[col] / cnt;
        const float ex2  = ws[DSZ + col] / cnt;
        const float var  = ex2 - mean * mean;
        const float rstd = rsqrtf(var + EPSV);
        const float g    = gamma[col] * rstd;
        const float bt   = beta[col];
        const float bb   = bias[col];
#pragma unroll
        for (int r = 0; r < 8; ++r)
            vals[r] = fmaxf(0.f, (x[r] + bb - mean) * g + bt);
    } else {
        const float f = features[(size_t)((b * DSZ + (col - DSZ)) * NSZ) + n];
#pragma unroll
        for (int r = 0; r < 8; ++r) vals[r] = f;
    }

    // out[b][c][n][k], c = col (0..7 encoded, 8..15 features), k = r + hi*8
    const size_t base = ((size_t)(b * 2 * DSZ + col) * NSZ + n) * KSZ + (size_t)hi * 8;
    float4* p = reinterpret_cast<float4*>(out + base);
    p[0] = make_float4(vals[0], vals[1], vals[2], vals[3]);
    p[1] = make_float4(vals[4], vals[5], vals[6], vals[7]);
}

extern "C" void kernel_launch(void* const* d_in, const int* in_sizes, int n_in,
                              void* d_out, int out_size, void* d_ws, size_t ws_size,
                              hipStream_t stream) {
    (void)in_sizes; (void)n_in; (void)out_size; (void)ws_size;
    const float* coords   = (const float*)d_in[0];
    const float* features = (const float*)d_in[1];
    const int*   idx      = (const int*)  d_in[2];
    const float* dist     = (const float*)d_in[3];
    const float* W        = (const float*)d_in[4];
    const float* bias     = (const float*)d_in[5];
    const float* gamma    = (const float*)d_in[6];
    const float* beta     = (const float*)d_in[7];
    float* out = (float*)d_out;
    float* ws  = (float*)d_ws;

    lse_zero_ws<<<1, 32, 0, stream>>>(ws);

    // Pass 1: 2048 blocks x 8 waves, grid-stride over 131072 (b,n) rows.
    lse_stats<<<2048, 256, 0, stream>>>(coords, idx, dist, W, bias, ws);

    // Pass 2: exactly one wave per (b,n): 131072 waves / 8 per block.
    lse_main<<<(BSZ * NSZ) / 8, 256, 0, stream>>>(coords, features, idx, dist,
                                                  W, bias, gamma, beta, ws, out);
}